// PyramidCRF_13091060318394
// MI455X (gfx1250) — compile-verified
//
#include <hip/hip_runtime.h>
#include <math.h>

// ---------------------------------------------------------------------------
// PyramidCRF on MI455X (gfx1250, wave32).
// CRF forward step rewritten as: p = exp(alpha - m); D = p @ exp(trans);
// alpha' = emit + m + log(D).   The 48x48 contraction runs on
// V_WMMA_F32_16X16X4_F32 (fp32 matrix core), 3 N-tiles x 12 K-chunks.
// loss_T (64 waves, latency-bound) additionally double-buffers the per-step
// emit tile into LDS with GLOBAL_LOAD_ASYNC_TO_LDS_B128 (ASYNCcnt pipeline).
// ---------------------------------------------------------------------------

typedef __attribute__((ext_vector_type(2))) float v2f;
typedef __attribute__((ext_vector_type(8))) float v8f;
typedef __attribute__((__vector_size__(16))) int int4v;   // builtin's b128 unit

#define P_LVL 16
#define B_DIM 64
#define T_DIM 512
#define N_TAG 48
#define NSTR  52      // padded LDS row stride: 16B-aligned rows + bank spread
#define KT    12      // 48 / 4  K-chunks
#define JT    3       // 48 / 16 N-tiles
#define WAVES 4       // waves per block (128 threads)

#if defined(__has_builtin)
# if __has_builtin(__builtin_amdgcn_global_load_async_to_lds_b128)
#  define HAS_ASYNC_LDS 1
# endif
#endif
#ifndef HAS_ASYNC_LDS
# define HAS_ASYNC_LDS 0
#endif

#if HAS_ASYNC_LDS
# if defined(__has_builtin) && __has_builtin(__builtin_amdgcn_s_wait_asynccnt)
#  define WAIT_ASYNC(n) __builtin_amdgcn_s_wait_asynccnt(n)
# else
#  define WAIT_ASYNC(n) asm volatile("s_wait_asynccnt %0" :: "i"(n) : "memory")
# endif
#else
# define WAIT_ASYNC(n) ((void)0)
#endif

// Stage one emit tile (16 rows x 48 floats = 192 x 16B chunks) into LDS.
// 6 x B128 transfers across 32 lanes; async when the toolchain exposes it.
__device__ __forceinline__ void stage_emit(const float* __restrict__ gbase,
                                           float (* __restrict__ dst)[NSTR],
                                           size_t seq_stride, int lane)
{
    #pragma unroll
    for (int i = 0; i < 6; ++i) {
        int chunk = i * 32 + lane;            // 0..191
        int s = chunk / 12, c = chunk % 12;
        const float* g = gbase + (size_t)s * seq_stride + (size_t)c * 4;
        float* l = &dst[s][c * 4];
#if HAS_ASYNC_LDS
        int4v* gg = (int4v*)const_cast<float*>(g);   // builtin wants non-const
        int4v* ll = (int4v*)l;
        __builtin_amdgcn_global_load_async_to_lds_b128(
            (__attribute__((address_space(1))) int4v*)gg,
            (__attribute__((address_space(3))) int4v*)ll, 0, 0);
#else
        *(float4*)l = *(const float4*)g;
#endif
    }
}

__device__ __forceinline__ float row_max48(const float* __restrict__ row)
{
    const float4* r4 = (const float4*)row;    // rows are 16B aligned (NSTR=52)
    float4 a = r4[0];
    float mx = fmaxf(fmaxf(a.x, a.y), fmaxf(a.z, a.w));
    #pragma unroll
    for (int k = 1; k < 12; ++k) {
        float4 b = r4[k];
        mx = fmaxf(mx, fmaxf(fmaxf(b.x, b.y), fmaxf(b.z, b.w)));
    }
    return mx;
}

// ---------------------------------------------------------------------------
// Kernel 1: loss_T. Sequences run along T (1024 seqs of length <=512).
// One wave owns 16 consecutive sequences (same pyramid level p -> uniform
// mask: valid steps are t < T - p).
// ---------------------------------------------------------------------------
__global__ __launch_bounds__(128) void crf_fwd_T(const float* __restrict__ logits,
                                                 const float* __restrict__ trans,
                                                 const int*   __restrict__ tags,
                                                 float* __restrict__ out)
{
    __shared__ __align__(16) float expT [N_TAG][NSTR];
    __shared__ __align__(16) float alpha[WAVES][16][NSTR];    // doubles as p (in-place exp)
    __shared__ __align__(16) float ebuf [WAVES][2][16][NSTR]; // emit double buffer
    __shared__ float mrow[WAVES][16];
    __shared__ float part[WAVES][16];

    const int tid  = threadIdx.x;
    const int lane = tid & 31;
    const int wave = tid >> 5;

    // block-shared exp(trans)
    for (int i = tid; i < N_TAG * N_TAG; i += blockDim.x) {
        expT[i / N_TAG][i % N_TAG] = __expf(trans[i]);
    }
    __syncthreads();

    const int wave_g = blockIdx.x * WAVES + wave;
    const int seq0   = wave_g * 16;            // first of 16 sequences
    const int p_lvl  = seq0 / B_DIM;           // uniform per wave
    const int Tvalid = T_DIM - p_lvl;          // mask: t < Tvalid

    const size_t seq_stride = (size_t)T_DIM * N_TAG;
    const float* lg0 = logits + (size_t)seq0 * seq_stride;

    // Preload B fragments: exp(trans) as 4x16 fp32 WMMA B tiles (step-invariant).
    // Layout: VGPR0 = {K=k0 (lanes 0-15) | K=k0+2 (lanes 16-31)}, VGPR1 = {k0+1 | k0+3}.
    v2f Bf[JT][KT];
    {
        const int khalf = (lane < 16) ? 0 : 2;
        const int n     = lane & 15;
        #pragma unroll
        for (int jt = 0; jt < JT; ++jt)
            #pragma unroll
            for (int k = 0; k < KT; ++k) {
                Bf[jt][k].x = expT[k * 4 + khalf + 0][jt * 16 + n];
                Bf[jt][k].y = expT[k * 4 + khalf + 1][jt * 16 + n];
            }
    }

    // alpha0 = logits[:, t=0, :]
    for (int i = lane; i < 16 * N_TAG; i += 32) {
        int s = i / N_TAG, j = i - s * N_TAG;
        alpha[wave][s][j] = lg0[(size_t)s * seq_stride + j];
    }

    // pre-stage emit tile for t=1 (Tvalid >= 497, always taken)
    stage_emit(lg0 + N_TAG, ebuf[wave][0], seq_stride, lane);

    for (int t = 1; t < Tvalid; ++t) {
        const int buf = (t - 1) & 1;

        // 1) per-sequence running max (12 x ds_load_b128 per row)
        if (lane < 16) mrow[wave][lane] = row_max48(&alpha[wave][lane][0]);

        // 2) p = exp(alpha - m), in place (p is dead after the WMMA consumes it)
        for (int i = lane; i < 16 * N_TAG; i += 32) {
            int s = i / N_TAG, j = i - s * N_TAG;
            alpha[wave][s][j] = __expf(alpha[wave][s][j] - mrow[wave][s]);
        }

        // 3) issue async staging of the NEXT emit tile -- overlaps the WMMA chain
        if (t + 1 < Tvalid)
            stage_emit(lg0 + (size_t)(t + 1) * N_TAG, ebuf[wave][buf ^ 1], seq_stride, lane);

        // 4) D = p @ exp(trans) -- 36x v_wmma_f32_16x16x4_f32, full EXEC
        v8f acc0 = {}; v8f acc1 = {}; v8f acc2 = {};
        {
            const int mr    = lane & 15;
            const int khalf = (lane < 16) ? 0 : 2;
            #pragma unroll
            for (int k = 0; k < KT; ++k) {
                v2f a;
                a.x = alpha[wave][mr][k * 4 + khalf + 0];
                a.y = alpha[wave][mr][k * 4 + khalf + 1];
                acc0 = __builtin_amdgcn_wmma_f32_16x16x4_f32(false, a, false, Bf[0][k],
                                                             (short)0, acc0, false, false);
                acc1 = __builtin_amdgcn_wmma_f32_16x16x4_f32(false, a, false, Bf[1][k],
                                                             (short)0, acc1, false, false);
                acc2 = __builtin_amdgcn_wmma_f32_16x16x4_f32(false, a, false, Bf[2][k],
                                                             (short)0, acc2, false, false);
            }
        }

        // 5) wait for the CURRENT buffer's async copies. Async loads complete
        //    in order: after issuing 6 new ops, <=6 outstanding means the 6
        //    older (current-buffer) ops are done. Last step: drain to 0.
        if (t + 1 < Tvalid) { WAIT_ASYNC(6); } else { WAIT_ASYNC(0); }

        // 6) alpha' = emit + m + log(D); C/D layout: VGPR r -> seq r (lanes 0-15)
        //    or seq r+8 (lanes 16-31), tag = tile*16 + (lane&15).
        {
            const int shalf = (lane < 16) ? 0 : 8;
            const int n     = lane & 15;
            #pragma unroll
            for (int r = 0; r < 8; ++r) {
                int s   = r + shalf;
                float m = mrow[wave][s];
                const float* es = &ebuf[wave][buf][s][0];
                alpha[wave][s][0 * 16 + n] = es[0 * 16 + n] + m + __logf(acc0[r]);
                alpha[wave][s][1 * 16 + n] = es[1 * 16 + n] + m + __logf(acc1[r]);
                alpha[wave][s][2 * 16 + n] = es[2 * 16 + n] + m + __logf(acc2[r]);
            }
        }
    }

    // log-partition + gold score per sequence
    if (lane < 16) {
        const int s = lane;
        float m = row_max48(&alpha[wave][s][0]);
        float sum = 0.f;
        for (int j = 0; j < N_TAG; ++j) sum += __expf(alpha[wave][s][j] - m);
        float log_den = m + __logf(sum);

        const float* lgs = lg0 + (size_t)s * seq_stride;
        const int*   tgs = tags + (size_t)(seq0 + s) * T_DIM;
        float score = 0.f;
        int   tprev = 0;
        for (int t = 0; t < T_DIM; ++t) {
            int tagc = tgs[t];
            if (t < Tvalid) {
                score += lgs[(size_t)t * N_TAG + tagc];           // emit (incl. last)
                if (t > 0) score += trans[tprev * N_TAG + tagc];  // transition
            }
            tprev = tagc;
        }
        part[wave][s] = log_den - score;
    }
    if (lane == 0) {
        float acc = 0.f;
        for (int s = 0; s < 16; ++s) acc += part[wave][s];
        atomicAdd(out, acc);
    }
}

// ---------------------------------------------------------------------------
// Kernel 2: loss_L (mode=0) / loss_R (mode=1). Sequences run along P
// (32768 seqs of length 16). Sequence s = t*B + b; a wave's 16 consecutive
// sequences share t -> masks uniform per wave. 2048 waves -> occupancy hides
// global latency, no async staging needed here.
//   mode 0 (L): element (s,p) at logits[p, b, t];      valid iff t <  T - p
//   mode 1 (R): element (s,p) at logits[p, b, (t-p)%T]; valid iff t >= p
// ---------------------------------------------------------------------------
__global__ __launch_bounds__(128) void crf_fwd_LR(const float* __restrict__ logits,
                                                  const float* __restrict__ trans,
                                                  const int*   __restrict__ tags,
                                                  float* __restrict__ out,
                                                  int mode)
{
    __shared__ __align__(16) float expT [N_TAG][NSTR];
    __shared__ __align__(16) float alpha[WAVES][16][NSTR];
    __shared__ float mrow[WAVES][16];
    __shared__ float part[WAVES][16];

    const int tid  = threadIdx.x;
    const int lane = tid & 31;
    const int wave = tid >> 5;

    for (int i = tid; i < N_TAG * N_TAG; i += blockDim.x) {
        expT[i / N_TAG][i % N_TAG] = __expf(trans[i]);
    }
    __syncthreads();

    const int wave_g = blockIdx.x * WAVES + wave;
    const int s0     = wave_g * 16;
    const int t      = s0 / B_DIM;     // uniform per wave
    const int b0     = s0 % B_DIM;

    const size_t lvl_stride = (size_t)T_DIM * N_TAG;   // stride between (p,b) rows

    v2f Bf[JT][KT];
    {
        const int khalf = (lane < 16) ? 0 : 2;
        const int n     = lane & 15;
        #pragma unroll
        for (int jt = 0; jt < JT; ++jt)
            #pragma unroll
            for (int k = 0; k < KT; ++k) {
                Bf[jt][k].x = expT[k * 4 + khalf + 0][jt * 16 + n];
                Bf[jt][k].y = expT[k * 4 + khalf + 1][jt * 16 + n];
            }
    }

    // alpha0: level p=0 (never rolled, never masked)
    for (int i = lane; i < 16 * N_TAG; i += 32) {
        int s = i / N_TAG, j = i - s * N_TAG;
        alpha[wave][s][j] = logits[((size_t)(b0 + s) * T_DIM + t) * N_TAG + j];
    }

    for (int p = 1; p < P_LVL; ++p) {
        const bool valid = mode ? (t >= p) : (t < T_DIM - p);   // uniform per wave
        if (!valid) continue;
        int tt = t;
        if (mode) { tt = t - p; if (tt < 0) tt += T_DIM; }
        const size_t baseP = ((size_t)(p * B_DIM + b0) * T_DIM + tt) * N_TAG;

        if (lane < 16) mrow[wave][lane] = row_max48(&alpha[wave][lane][0]);

        for (int i = lane; i < 16 * N_TAG; i += 32) {
            int s = i / N_TAG, j = i - s * N_TAG;
            alpha[wave][s][j] = __expf(alpha[wave][s][j] - mrow[wave][s]);
        }

        v8f acc0 = {}; v8f acc1 = {}; v8f acc2 = {};
        {
            const int mr    = lane & 15;
            const int khalf = (lane < 16) ? 0 : 2;
            #pragma unroll
            for (int k = 0; k < KT; ++k) {
                v2f a;
                a.x = alpha[wave][mr][k * 4 + khalf + 0];
                a.y = alpha[wave][mr][k * 4 + khalf + 1];
                acc0 = __builtin_amdgcn_wmma_f32_16x16x4_f32(false, a, false, Bf[0][k],
                                                             (short)0, acc0, false, false);
                acc1 = __builtin_amdgcn_wmma_f32_16x16x4_f32(false, a, false, Bf[1][k],
                                                             (short)0, acc1, false, false);
                acc2 = __builtin_amdgcn_wmma_f32_16x16x4_f32(false, a, false, Bf[2][k],
                                                             (short)0, acc2, false, false);
            }
        }
        {
            const int shalf = (lane < 16) ? 0 : 8;
            const int n     = lane & 15;
            #pragma unroll
            for (int r = 0; r < 8; ++r) {
                int s   = r + shalf;
                float m = mrow[wave][s];
                const float* es = logits + baseP + (size_t)s * lvl_stride;
                alpha[wave][s][0 * 16 + n] = es[0 * 16 + n] + m + __logf(acc0[r]);
                alpha[wave][s][1 * 16 + n] = es[1 * 16 + n] + m + __logf(acc1[r]);
                alpha[wave][s][2 * 16 + n] = es[2 * 16 + n] + m + __logf(acc2[r]);
            }
        }
    }

    if (lane < 16) {
        const int s = lane;
        const int b = b0 + s;
        float m = row_max48(&alpha[wave][s][0]);
        float sum = 0.f;
        for (int j = 0; j < N_TAG; ++j) sum += __expf(alpha[wave][s][j] - m);
        float log_den = m + __logf(sum);

        float score = 0.f;
        int   tprev = 0;
        for (int p = 0; p < P_LVL; ++p) {
            bool valid = mode ? (t >= p) : (t < T_DIM - p);
            int tt = t;
            if (mode) { tt = t - p; if (tt < 0) tt += T_DIM; }
            size_t row  = (size_t)(p * B_DIM + b) * T_DIM + tt;
            int    tagc = tags[row];
            if (valid) {
                score += logits[row * N_TAG + tagc];              // emit (incl. last)
                if (p > 0) score += trans[tprev * N_TAG + tagc];  // transition
            }
            tprev = tagc;
        }
        part[wave][s] = log_den - score;
    }
    if (lane == 0) {
        float acc = 0.f;
        for (int s = 0; s < 16; ++s) acc += part[wave][s];
        atomicAdd(out, acc);
    }
}

__global__ void zero_out_kernel(float* __restrict__ out) {
    if (threadIdx.x < 3) out[threadIdx.x] = 0.f;
}

extern "C" void kernel_launch(void* const* d_in, const int* in_sizes, int n_in,
                              void* d_out, int out_size, void* d_ws, size_t ws_size,
                              hipStream_t stream) {
    (void)in_sizes; (void)n_in; (void)out_size; (void)d_ws; (void)ws_size;
    const float* logits  = (const float*)d_in[0];
    const float* trans_T = (const float*)d_in[1];
    const float* trans_L = (const float*)d_in[2];
    const float* trans_R = (const float*)d_in[3];
    const int*   tags    = (const int*)  d_in[4];
    float* out = (float*)d_out;

    zero_out_kernel<<<1, 64, 0, stream>>>(out);

    // loss_T: 1024 sequences / 16 per wave / 4 waves per block = 16 blocks
    crf_fwd_T<<<16, 128, 0, stream>>>(logits, trans_T, tags, out + 0);

    // loss_L / loss_R: 32768 sequences / 16 per wave / 4 waves per block = 512 blocks
    crf_fwd_LR<<<512, 128, 0, stream>>>(logits, trans_L, tags, out + 1, 0);
    crf_fwd_LR<<<512, 128, 0, stream>>>(logits, trans_R, tags, out + 2, 1);
}